// MinkResNet_53231824667060
// MI455X (gfx1250) — compile-verified
//
#include <hip/hip_runtime.h>
#include <hip/hip_bf16.h>
#include <math.h>

typedef __attribute__((ext_vector_type(2))) float v2f;
typedef __attribute__((ext_vector_type(8))) float v8f;

// ---------------------------------------------------------------------------
// Copy x (N0 x 1) into a padded buffer with an appended zero row.
// ---------------------------------------------------------------------------
__global__ __launch_bounds__(256)
void pad_copy_kernel(const float* __restrict__ x, float* __restrict__ xp, int n) {
  int i = blockIdx.x * 256 + threadIdx.x;
  if (i <= n) xp[i] = (i < n) ? x[i] : 0.0f;
}

// ---------------------------------------------------------------------------
// Sparse conv, one kernel offset k per launch (deterministic: each output
// row is touched by at most one lane within a launch -> plain RMW scatter).
//
// One wave per block computes a 64(M: map entries) x 16(N: out channels)
// tile as 4 register-blocked M-subtiles sharing one B fragment per K-step,
// accumulating over ci in steps of 4 via V_WMMA_F32_16X16X4_F32 (fp32 in,
// fp32 accum — matches the fp32 reference).
//
// Fragment layouts (ISA 7.12.2), lane = (l, hl):
//   A 16x4 :  lane holds A[l][2*hl + j], j=0,1  (aligned float2 gather)
//   B 4x16 :  lane holds B[2*hl + j][l]
//   C/D    :  VGPR v holds D[v + 8*hl][l]
// ---------------------------------------------------------------------------
#define MT 4   // M-subtiles per wave (64 rows)

__global__ __launch_bounds__(32)
void sparse_conv_wmma(const float* __restrict__ xin,     // (N0+1) x ci, rows >= n_in are zero
                      const float* __restrict__ W,       // 27 x ci x co
                      const int*   __restrict__ in_map,  // 27 x Emax (pad = n_in)
                      const int*   __restrict__ out_map, // 27 x Emax (pad = n_out)
                      float*       __restrict__ out,     // (N0+1) x co, pre-zeroed
                      const int*   __restrict__ n_out_ptr,
                      int k, int Emax, int ci, int co, int zero_row) {
  const int lane = threadIdx.x & 31;
  const int hl   = lane >> 4;   // half: 0 or 1
  const int l    = lane & 15;
  const int e0   = blockIdx.x * (16 * MT);
  const int n0   = blockIdx.y * 16;
  const int n_out = *n_out_ptr;
  const int koff = 2 * hl;

  // Gather row pointers for the 4 M-subtiles (both halves compute the same).
  const float* arow[MT];
  #pragma unroll
  for (int t = 0; t < MT; ++t) {
    const int e = e0 + t * 16 + l;
    int in_idx = zero_row;                     // rows beyond Emax read the zero row
    if (e < Emax) in_idx = in_map[(size_t)k * Emax + e];
    if (in_idx > zero_row) in_idx = zero_row;  // safety clamp (pad == n_in <= zero_row)
    arow[t] = xin + (size_t)in_idx * (size_t)ci;
  }

  const float* __restrict__ wbase = W + (size_t)k * (size_t)ci * (size_t)co + n0 + l;

  v8f acc[MT];
  #pragma unroll
  for (int t = 0; t < MT; ++t) acc[t] = (v8f){};

  if ((ci & 3) == 0) {
    // Clean path: ci is a multiple of 4 (all layers except the first).
    #pragma unroll 2
    for (int kk = 0; kk < ci; kk += 4) {
      const int k0 = kk + koff;
      v2f b;
      b.x = wbase[(size_t)k0 * co];
      b.y = wbase[(size_t)(k0 + 1) * co];
      #pragma unroll
      for (int t = 0; t < MT; ++t) {
        v2f a = *(const v2f*)(arow[t] + k0);   // 8B aligned (ci%4==0, k0 even)
        acc[t] = __builtin_amdgcn_wmma_f32_16x16x4_f32(
            false, a, false, b, (short)0, acc[t], false, false);
      }
    }
  } else {
    // Guarded path (ci == 1 in the first layer): zero out-of-range K slots.
    for (int kk = 0; kk < ci; kk += 4) {
      const int k0 = kk + koff;
      const bool v0 = (k0 < ci);
      const bool v1 = (k0 + 1 < ci);
      v2f b;
      b.x = v0 ? wbase[(size_t)k0 * co]       : 0.0f;
      b.y = v1 ? wbase[(size_t)(k0 + 1) * co] : 0.0f;
      #pragma unroll
      for (int t = 0; t < MT; ++t) {
        v2f a;
        a.x = v0 ? arow[t][k0]     : 0.0f;
        a.y = v1 ? arow[t][k0 + 1] : 0.0f;
        acc[t] = __builtin_amdgcn_wmma_f32_16x16x4_f32(
            false, a, false, b, (short)0, acc[t], false, false);
      }
    }
  }

  // Scatter-accumulate: lane owns D[v + 8*hl][l] of each subtile; exclusive
  // row ownership within this launch -> plain read-modify-write, deterministic.
  #pragma unroll
  for (int t = 0; t < MT; ++t) {
    #pragma unroll
    for (int v = 0; v < 8; ++v) {
      const int er = e0 + t * 16 + v + 8 * hl;
      if (er < Emax) {
        const int oidx = out_map[(size_t)k * Emax + er];
        if (oidx < n_out) {
          float* p = out + (size_t)oidx * co + n0 + l;
          *p += acc[t][v];
        }
      }
    }
  }
}

// ---------------------------------------------------------------------------
// BN stats: per-channel mean / biased var over the n_out active rows.
// Block = 64 channels x 4 row-groups; fixed-order reduction (deterministic).
// ---------------------------------------------------------------------------
__global__ __launch_bounds__(256)
void bn_stats_kernel(const float* __restrict__ h, const int* __restrict__ n_out_ptr,
                     int co, float* __restrict__ meanv, float* __restrict__ varv) {
  __shared__ float s_sum[256];
  __shared__ float s_sq[256];
  const int n = *n_out_ptr;
  const int c = blockIdx.x * 64 + (threadIdx.x & 63);
  const int g = threadIdx.x >> 6;  // 0..3
  float s = 0.0f, q = 0.0f;
  for (int r = g; r < n; r += 4) {
    float v = h[(size_t)r * co + c];
    s += v;
    q += v * v;
  }
  s_sum[threadIdx.x] = s;
  s_sq[threadIdx.x]  = q;
  __syncthreads();
  if (g == 0) {
    const int t = threadIdx.x;
    float ts = s_sum[t] + s_sum[t + 64] + s_sum[t + 128] + s_sum[t + 192];
    float tq = s_sq[t]  + s_sq[t + 64]  + s_sq[t + 128]  + s_sq[t + 192];
    float inv = 1.0f / (float)n;
    float m = ts * inv;
    meanv[c] = m;
    varv[c]  = tq * inv - m * m;
  }
}

// ---------------------------------------------------------------------------
// BN apply + ReLU, in place over rows [0, n_out). Rows beyond n_out stay zero
// (required for the pad-row gather trick in the next layer).
// ---------------------------------------------------------------------------
__global__ __launch_bounds__(256)
void bn_apply_kernel(float* __restrict__ h, const int* __restrict__ n_out_ptr,
                     int co, int comask,
                     const float* __restrict__ gamma, const float* __restrict__ beta,
                     const float* __restrict__ meanv, const float* __restrict__ varv) {
  const int n = *n_out_ptr;
  size_t i = (size_t)blockIdx.x * 256 + threadIdx.x;
  size_t total = (size_t)n * (size_t)co;
  if (i >= total) return;
  const int c = (int)(i & (size_t)comask);   // co is a power of two
  float v = h[i];
  float y = (v - meanv[c]) * rsqrtf(varv[c] + 1e-5f) * gamma[c] + beta[c];
  h[i] = fmaxf(y, 0.0f);
}

// ---------------------------------------------------------------------------
// GeM pooling: per-batch element counts (deterministic tree reduction).
// ---------------------------------------------------------------------------
__global__ __launch_bounds__(256)
void gem_count_kernel(const int* __restrict__ bidx, int n, int nb,
                      float* __restrict__ cnts) {
  __shared__ float s[256];
  for (int b = 0; b < nb; ++b) {
    float c = 0.0f;
    for (int r = threadIdx.x; r < n; r += 256) c += (bidx[r] == b) ? 1.0f : 0.0f;
    s[threadIdx.x] = c;
    __syncthreads();
    for (int o = 128; o > 0; o >>= 1) {
      if (threadIdx.x < (unsigned)o) s[threadIdx.x] += s[threadIdx.x + o];
      __syncthreads();
    }
    if (threadIdx.x == 0) cnts[b] = s[0];
    __syncthreads();
  }
}

// ---------------------------------------------------------------------------
// GeM sums: per (batch, channel) sum of clamp(h,eps)^p. Fixed-order reduce.
// ---------------------------------------------------------------------------
__global__ __launch_bounds__(256)
void gem_sum_kernel(const float* __restrict__ h, const int* __restrict__ bidx,
                    int n, int co, const float* __restrict__ p_ptr, int nb,
                    float* __restrict__ sums) {
  __shared__ float s[256];
  const float p = *p_ptr;
  const int c = blockIdx.x * 64 + (threadIdx.x & 63);
  const int g = threadIdx.x >> 6;
  for (int b = 0; b < nb; ++b) {
    float acc = 0.0f;
    for (int r = g; r < n; r += 4) {
      if (bidx[r] == b) {
        float v = fmaxf(h[(size_t)r * co + c], 1e-6f);
        acc += powf(v, p);
      }
    }
    s[threadIdx.x] = acc;
    __syncthreads();
    if (g == 0) {
      const int t = threadIdx.x;
      sums[(size_t)b * co + c] = s[t] + s[t + 64] + s[t + 128] + s[t + 192];
    }
    __syncthreads();
  }
}

__global__ __launch_bounds__(256)
void gem_final_kernel(const float* __restrict__ sums, const float* __restrict__ cnts,
                      const float* __restrict__ p_ptr, float* __restrict__ out,
                      int total, int co) {
  int i = blockIdx.x * 256 + threadIdx.x;
  if (i >= total) return;
  const int b = i / co;
  const float p = *p_ptr;
  out[i] = powf(sums[i] / cnts[b], 1.0f / p);
}

// ---------------------------------------------------------------------------
// Orchestration.
// d_in layout (setup_inputs dict order, lists flattened):
//   [0] x, [1..9] weights, [10..18] gammas, [19..27] betas, [28] p,
//   [29..37] in_maps, [38..46] out_maps, [47..55] n_outs (1-elem each),
//   [56] batch_idx, [57] n_batch
// ---------------------------------------------------------------------------
extern "C" void kernel_launch(void* const* d_in, const int* in_sizes, int n_in,
                              void* d_out, int out_size, void* d_ws, size_t ws_size,
                              hipStream_t stream) {
  static const int CIs[9] = {1, 64, 128, 128, 256, 256, 512, 512, 1024};
  static const int COs[9] = {64, 128, 128, 256, 256, 512, 512, 1024, 1024};

  const float* x = (const float*)d_in[0];
  const float* Wt[9]; const float* ga[9]; const float* be[9];
  const int *inm[9], *outm[9], *nop[9];
  int Emax[9];
  for (int i = 0; i < 9; ++i) {
    Wt[i]  = (const float*)d_in[1 + i];
    ga[i]  = (const float*)d_in[10 + i];
    be[i]  = (const float*)d_in[19 + i];
    inm[i] = (const int*)d_in[29 + i];
    outm[i]= (const int*)d_in[38 + i];
    nop[i] = (const int*)d_in[47 + i];
    Emax[i]= in_sizes[29 + i] / 27;
  }
  const float* p      = (const float*)d_in[28];
  const int*   bidx   = (const int*)d_in[56];
  const int    Nlast  = in_sizes[56];          // exact n_out of last layer
  const int    N0     = in_sizes[0];           // x is (N0, 1)
  const int    B      = out_size / 1024;       // output is (B, 1024)

  // Workspace layout (floats).
  float* ws = (float*)d_ws;
  const size_t rowcap = (size_t)N0 + 1;
  float* bufA  = ws;
  float* bufB  = bufA + rowcap * 1024;
  float* xpad  = bufB + rowcap * 1024;
  float* meanv = xpad + ((rowcap + 63) & ~(size_t)63);
  float* varv  = meanv + 1024;
  float* sums  = varv + 1024;
  float* cnts  = sums + 16 * 1024;
  (void)ws_size; (void)n_in;

  // x -> padded buffer with zero row at index N0.
  pad_copy_kernel<<<(unsigned)((N0 + 256) / 256), 256, 0, stream>>>(x, xpad, N0);

  const float* cur = xpad;
  float* nxt = bufA;
  for (int i = 0; i < 9; ++i) {
    // Zero the whole output buffer: makes rows >= n_out implicit zero pad rows.
    hipMemsetAsync(nxt, 0, rowcap * (size_t)COs[i] * sizeof(float), stream);

    dim3 grid((unsigned)((Emax[i] + 16 * MT - 1) / (16 * MT)),
              (unsigned)(COs[i] / 16));
    for (int k = 0; k < 27; ++k) {
      sparse_conv_wmma<<<grid, 32, 0, stream>>>(
          cur, Wt[i], inm[i], outm[i], nxt, nop[i],
          k, Emax[i], CIs[i], COs[i], N0);
    }

    bn_stats_kernel<<<(unsigned)(COs[i] / 64), 256, 0, stream>>>(
        nxt, nop[i], COs[i], meanv, varv);

    size_t tot = (size_t)N0 * (size_t)COs[i];
    bn_apply_kernel<<<(unsigned)((tot + 255) / 256), 256, 0, stream>>>(
        nxt, nop[i], COs[i], COs[i] - 1, ga[i], be[i], meanv, varv);

    cur = nxt;
    nxt = (nxt == bufA) ? bufB : bufA;
  }

  gem_count_kernel<<<1, 256, 0, stream>>>(bidx, Nlast, B, cnts);
  gem_sum_kernel<<<1024 / 64, 256, 0, stream>>>(cur, bidx, Nlast, 1024, p, B, sums);
  gem_final_kernel<<<(unsigned)((B * 1024 + 255) / 256), 256, 0, stream>>>(
      sums, cnts, p, (float*)d_out, B * 1024, 1024);
}